// GCN_13125420056951
// MI455X (gfx1250) — compile-verified
//
#include <hip/hip_runtime.h>
#include <math.h>
#include <stdint.h>

// ---------------------------------------------------------------------------
// GCN (2-layer, GCNConv + ReLU + GCNConv + log_softmax) for MI455X / gfx1250.
//
// Design notes (compile-only reasoning):
//  * The workload is scatter-bound. Gather/scatter targets xw1 (6.4MB) and
//    xw2 (8MB) fit in the 192MB L2, so edge-parallel float atomics are the
//    right tool; HBM traffic is ~150-200MB -> ~10us at 23.3 TB/s.
//  * Dense GEMMs are tiny (0.47 GFLOP total) but are done on the WMMA pipe
//    with V_WMMA_F32_16X16X4_F32: fp32-exact, 16-node tiles, wave32.
//  * Fragment layouts per CDNA5 ISA 7.12.2:
//      A (16x4 f32): lanes 0-15 -> K=0,1 in v[0],v[1]; lanes 16-31 -> K=2,3.
//      B (4x16 f32): assumed mirror layout (K rows striped by lane-half).
//      C/D (16x16 f32): VGPR r -> M=r (lanes 0-15) / M=r+8 (lanes 16-31).
// ---------------------------------------------------------------------------

typedef float v2f __attribute__((ext_vector_type(2)));
typedef float v8f __attribute__((ext_vector_type(8)));

#define F_IN  128
#define F_HID 16
#define F_OUT 20
#define TPB   256

// ----------------------------- utility kernels -----------------------------

__global__ void gcn_fill_f32(float* __restrict__ p, float v, int n) {
  int i = blockIdx.x * blockDim.x + threadIdx.x;
  if (i < n) p[i] = v;
}

// deg[dst] += 1 per edge (deg buffer pre-filled with 1.0 for self-loops)
__global__ void gcn_deg(const int* __restrict__ dst, float* __restrict__ deg, int E) {
  int e = blockIdx.x * blockDim.x + threadIdx.x;
  if (e < E) atomicAdd(&deg[dst[e]], 1.0f);
}

// deg -> dinv in place
__global__ void gcn_dinv(float* __restrict__ deg, int n) {
  int i = blockIdx.x * blockDim.x + threadIdx.x;
  if (i < n) {
    float d = deg[i];
    deg[i] = (d > 0.0f) ? rsqrtf(d) : 0.0f;
  }
}

// norm[e] = dinv[src[e]] * dinv[dst[e]]
__global__ void gcn_norm(const int* __restrict__ src, const int* __restrict__ dst,
                         const float* __restrict__ dinv, float* __restrict__ norm, int E) {
  int e = blockIdx.x * blockDim.x + threadIdx.x;
  if (e < E) norm[e] = dinv[src[e]] * dinv[dst[e]];
}

// ------------------------------- WMMA GEMMs --------------------------------
// xw1[N,16] = x[N,128] @ W1[128,16]; one wave32 per 16-node tile.
__global__ void gcn_gemm1_wmma(const float* __restrict__ x, const float* __restrict__ W,
                               float* __restrict__ xw, int ntiles) {
  int wave = (blockIdx.x * blockDim.x + threadIdx.x) >> 5;
  int lane = threadIdx.x & 31;
  if (wave >= ntiles) return;               // wave-uniform: EXEC stays all-ones
  int m    = lane & 15;                     // row within tile (also col n for B)
  int koff = (lane >> 4) << 1;              // 0 for lanes 0-15, 2 for lanes 16-31
  const float* xrow = x + (size_t)(wave * 16 + m) * F_IN;
  v8f c = {};
  #pragma unroll 4
  for (int k0 = 0; k0 < F_IN; k0 += 4) {
    v2f a, b;
    a.x = xrow[k0 + koff];
    a.y = xrow[k0 + koff + 1];
    b.x = W[(k0 + koff) * F_HID + m];
    b.y = W[(k0 + koff + 1) * F_HID + m];
    c = __builtin_amdgcn_wmma_f32_16x16x4_f32(false, a, false, b, (short)0, c,
                                              false, false);
  }
  int rbase = (lane >> 4) << 3;             // M offset: 0 or 8
  float* ob = xw + (size_t)(wave * 16) * F_HID;
  #pragma unroll
  for (int r = 0; r < 8; ++r)
    ob[(r + rbase) * F_HID + m] = c[r];     // coalesced 64B per half-wave
}

// xw2[N,20] = h1[N,16] @ W2[16,20]; K=16 (4 steps), N padded to 2x16 halves.
__global__ void gcn_gemm2_wmma(const float* __restrict__ h, const float* __restrict__ W,
                               float* __restrict__ xw, int ntiles) {
  int wave = (blockIdx.x * blockDim.x + threadIdx.x) >> 5;
  int lane = threadIdx.x & 31;
  if (wave >= ntiles) return;
  int m    = lane & 15;
  int koff = (lane >> 4) << 1;
  int n1   = 16 + m;                        // column in second N-half (valid < 20)
  const float* hrow = h + (size_t)(wave * 16 + m) * F_HID;
  v8f c0 = {}, c1 = {};
  #pragma unroll
  for (int k0 = 0; k0 < F_HID; k0 += 4) {
    v2f a;
    a.x = hrow[k0 + koff];
    a.y = hrow[k0 + koff + 1];
    v2f b0, b1;
    b0.x = W[(k0 + koff) * F_OUT + m];
    b0.y = W[(k0 + koff + 1) * F_OUT + m];
    b1.x = (n1 < F_OUT) ? W[(k0 + koff) * F_OUT + n1] : 0.0f;
    b1.y = (n1 < F_OUT) ? W[(k0 + koff + 1) * F_OUT + n1] : 0.0f;
    c0 = __builtin_amdgcn_wmma_f32_16x16x4_f32(false, a, false, b0, (short)0, c0,
                                               false, false);
    c1 = __builtin_amdgcn_wmma_f32_16x16x4_f32(false, a, false, b1, (short)0, c1,
                                               false, false);
  }
  int rbase = (lane >> 4) << 3;
  float* ob = xw + (size_t)(wave * 16) * F_OUT;
  #pragma unroll
  for (int r = 0; r < 8; ++r) {
    ob[(r + rbase) * F_OUT + m] = c0[r];
    if (n1 < F_OUT) ob[(r + rbase) * F_OUT + n1] = c1[r];
  }
}

// --------------------------- edge scatter kernels --------------------------
// 16 lanes per edge: agg1[dst,f] += xw1[src,f] * norm[e]   (L2-resident targets)
__global__ void gcn_agg16(const int* __restrict__ src, const int* __restrict__ dst,
                          const float* __restrict__ norm, const float* __restrict__ xw,
                          float* __restrict__ agg, int E) {
  int idx = blockIdx.x * blockDim.x + threadIdx.x;
  if (idx >= E * F_HID) return;
  int e = idx >> 4;
  int f = idx & 15;
  int s = src[e], d = dst[e];
  float v = xw[(size_t)s * F_HID + f] * norm[e];
  atomicAdd(&agg[(size_t)d * F_HID + f], v);
}

// 20 lanes per edge: agg2[dst,f] += xw2[src,f] * norm[e]
__global__ void gcn_agg20(const int* __restrict__ src, const int* __restrict__ dst,
                          const float* __restrict__ norm, const float* __restrict__ xw,
                          float* __restrict__ agg, int E) {
  int idx = blockIdx.x * blockDim.x + threadIdx.x;
  if (idx >= E * F_OUT) return;
  int e = idx / F_OUT;
  int f = idx - e * F_OUT;
  int s = src[e], d = dst[e];
  float v = xw[(size_t)s * F_OUT + f] * norm[e];
  atomicAdd(&agg[(size_t)d * F_OUT + f], v);
}

// h1 = relu(agg1 + xw1 * dinv^2 (self-loop) + b1), in place over agg1
__global__ void gcn_selfbias_relu(float* __restrict__ agg, const float* __restrict__ xw,
                                  const float* __restrict__ dinv, const float* __restrict__ b,
                                  int N) {
  int idx = blockIdx.x * blockDim.x + threadIdx.x;
  if (idx >= N * F_HID) return;
  int i = idx >> 4;
  int f = idx & 15;
  float di = dinv[i];
  float v = agg[idx] + xw[idx] * di * di + b[f];
  agg[idx] = fmaxf(v, 0.0f);
}

// out = log_softmax(agg2 + xw2 * dinv^2 + b2) per node
__global__ void gcn_final(const float* __restrict__ agg, const float* __restrict__ xw,
                          const float* __restrict__ dinv, const float* __restrict__ b,
                          float* __restrict__ out, int N) {
  int i = blockIdx.x * blockDim.x + threadIdx.x;
  if (i >= N) return;
  float d2 = dinv[i] * dinv[i];
  float v[F_OUT];
  float mx = -INFINITY;
  #pragma unroll
  for (int f = 0; f < F_OUT; ++f) {
    size_t o = (size_t)i * F_OUT + f;
    v[f] = agg[o] + xw[o] * d2 + b[f];
    mx = fmaxf(mx, v[f]);
  }
  float s = 0.0f;
  #pragma unroll
  for (int f = 0; f < F_OUT; ++f) s += expf(v[f] - mx);
  float lse = mx + logf(s);
  #pragma unroll
  for (int f = 0; f < F_OUT; ++f) out[(size_t)i * F_OUT + f] = v[f] - lse;
}

// --------------------------------- launch ----------------------------------

extern "C" void kernel_launch(void* const* d_in, const int* in_sizes, int n_in,
                              void* d_out, int out_size, void* d_ws, size_t ws_size,
                              hipStream_t stream) {
  const float* x  = (const float*)d_in[0];
  const int*   ei = (const int*)d_in[1];     // edge_index flat [2, E] (int)
  const float* W1 = (const float*)d_in[2];
  const float* b1 = (const float*)d_in[3];
  const float* W2 = (const float*)d_in[4];
  const float* b2 = (const float*)d_in[5];
  float* out = (float*)d_out;

  const int N = in_sizes[0] / F_IN;          // 100000
  const int E = in_sizes[1] / 2;             // 3200000
  const int* src = ei;
  const int* dst = ei + E;

  // workspace carve-up (floats): dinv[N] norm[E] xw1[16N] h1[16N] xw2[20N] agg2[20N]
  float* dinv = (float*)d_ws;
  float* norm = dinv + N;
  float* xw1  = norm + E;
  float* h1   = xw1 + (size_t)N * F_HID;     // agg1 -> relu'd h1 in place
  float* xw2  = h1  + (size_t)N * F_HID;
  float* agg2 = xw2 + (size_t)N * F_OUT;

  auto blocks = [](long long total) { return (int)((total + TPB - 1) / TPB); };

  // degree (init 1.0 for self-loop), dinv, per-edge norm
  gcn_fill_f32<<<blocks(N), TPB, 0, stream>>>(dinv, 1.0f, N);
  gcn_deg<<<blocks(E), TPB, 0, stream>>>(dst, dinv, E);
  gcn_dinv<<<blocks(N), TPB, 0, stream>>>(dinv, N);
  gcn_norm<<<blocks(E), TPB, 0, stream>>>(src, dst, dinv, norm, E);

  // layer 1: WMMA GEMM, zero accumulator, scatter, self-loop+bias+relu
  const int ntiles = N / 16;                 // N is a multiple of 16
  gcn_gemm1_wmma<<<blocks((long long)ntiles * 32), TPB, 0, stream>>>(x, W1, xw1, ntiles);
  gcn_fill_f32<<<blocks((long long)N * F_HID), TPB, 0, stream>>>(h1, 0.0f, N * F_HID);
  gcn_agg16<<<blocks((long long)E * F_HID), TPB, 0, stream>>>(src, dst, norm, xw1, h1, E);
  gcn_selfbias_relu<<<blocks((long long)N * F_HID), TPB, 0, stream>>>(h1, xw1, dinv, b1, N);

  // layer 2: WMMA GEMM (padded N), zero accumulator, scatter, log_softmax
  gcn_gemm2_wmma<<<blocks((long long)ntiles * 32), TPB, 0, stream>>>(h1, W2, xw2, ntiles);
  gcn_fill_f32<<<blocks((long long)N * F_OUT), TPB, 0, stream>>>(agg2, 0.0f, N * F_OUT);
  gcn_agg20<<<blocks((long long)E * F_OUT), TPB, 0, stream>>>(src, dst, norm, xw2, agg2, E);
  gcn_final<<<blocks(N), TPB, 0, stream>>>(agg2, xw2, dinv, b2, out, N);
}